// BAR_86045374808446
// MI455X (gfx1250) — compile-verified
//
#include <hip/hip_runtime.h>

typedef __attribute__((ext_vector_type(2))) float v2f;
typedef __attribute__((ext_vector_type(8))) float v8f;

#define EPSV   1e-5f
#define LOG2E  1.4426950408889634f

// ---------------------------------------------------------------------------
// Pass 1: per-row {sum, sumsq} over d for every (t,h) row of A and B.
// Memory row index w: offset w*64 in A/B (w = t*H + h).  One wave per row.
// ---------------------------------------------------------------------------
__global__ __launch_bounds__(256) void row_stats(
    const float* __restrict__ A, const float* __restrict__ B,
    v2f* __restrict__ WA, v2f* __restrict__ WB, int nrows)
{
    const int lane = threadIdx.x & 31;
    const int w    = blockIdx.x * 8 + (threadIdx.x >> 5);
    if (w >= nrows) return;

    const v2f xa = *(const v2f*)(A + w * 64 + 2 * lane);
    const v2f xb = *(const v2f*)(B + w * 64 + 2 * lane);
    v2f sa = { xa.x + xa.y, xa.x * xa.x + xa.y * xa.y };
    v2f sb = { xb.x + xb.y, xb.x * xb.x + xb.y * xb.y };
#pragma unroll
    for (int off = 16; off >= 1; off >>= 1) {
        v2f oa, ob;
        oa.x = __shfl_xor(sa.x, off, 32);
        oa.y = __shfl_xor(sa.y, off, 32);
        ob.x = __shfl_xor(sb.x, off, 32);
        ob.y = __shfl_xor(sb.y, off, 32);
        sa += oa;                 // v_pk_add_f32
        sb += ob;
    }
    if (lane == 0) { WA[w] = sa; WB[w] = sb; }
}

// ---------------------------------------------------------------------------
// Pass 2: one wave per (head, 16-row i-tile).
// RS = H*D = 512 floats row stride; lane owns d = {2*lane, 2*lane+1}.
// ---------------------------------------------------------------------------
__global__ __launch_bounds__(32) void pairnorm_exp_causal(
    const float* __restrict__ A, const float* __restrict__ B,
    const v2f* __restrict__ WA, const v2f* __restrict__ WB,
    float* __restrict__ O, int T, int NT)
{
    const int D = 64, RS = 512, H = 8;

    const int h    = blockIdx.x / NT;
    const int it   = blockIdx.x % NT;
    const int i0   = it * 16;
    const int lane = threadIdx.x;
    const int hf   = lane >> 4;    // 0 or 1 (lane half)
    const int m    = lane & 15;

    __shared__ float rv_l[256];    // rsqrt(var+eps)*log2e per pair
    __shared__ float nmr_l[256];   // -mu*rv per pair

    const float* Ab = A + h * D;
    const float* Bb = B + h * D;

    // ---- A-tile values (contiguous float2 per lane) ----
    v2f a_val[16], acc[16];
#pragma unroll
    for (int r = 0; r < 16; ++r) {
        a_val[r] = *(const v2f*)(Ab + (i0 + r) * RS + 2 * lane);
        acc[r]   = (v2f){0.f, 0.f};
    }

    // ---- A-row stats for this lane's 8 C-rows (row = v + 8*hf) ----
    v2f asq[8];
#pragma unroll
    for (int v = 0; v < 8; ++v) {
        asq[v] = WA[(i0 + v + 8 * hf) * H + h];
    }

    // ---- WMMA A fragments: 16x4 f32, K-step k covers d = 4k..4k+3.
    // lanes 0-15 = M, VGPR0 = K(2*hf), VGPR1 = K(2*hf+1).
    v2f afrag[16];
#pragma unroll
    for (int k = 0; k < 16; ++k) {
        afrag[k] = *(const v2f*)(Ab + (i0 + m) * RS + 4 * k + 2 * hf);
    }

    for (int jt = 0; jt <= it; ++jt) {
        const int j0 = jt * 16;

        // ---- B-tile values ----
        v2f b_val[16];
#pragma unroll
        for (int r = 0; r < 16; ++r) {
            b_val[r] = *(const v2f*)(Bb + (j0 + r) * RS + 2 * lane);
        }
        // B-row stats for this lane's column (j = j0 + m)
        const v2f bsq = WB[(j0 + m) * H + h];

        // ---- Cross-term tile C[i,j] = sum_d a[i,d]*b[j,d] via f32 WMMA ----
        v8f c = {};
#pragma unroll
        for (int k = 0; k < 16; ++k) {
            v2f bf = *(const v2f*)(Bb + (j0 + m) * RS + 4 * k + 2 * hf);
            c = __builtin_amdgcn_wmma_f32_16x16x4_f32(
                    false, afrag[k], false, bf, (short)0, c, false, false);
        }

        // ---- Per-pair constants.  C/D layout: c[v] = C[row v+8*hf, col m] ----
#pragma unroll
        for (int v = 0; v < 8; ++v) {
            const int r     = v + 8 * hf;
            const float mu  = (asq[v].x + bsq.x) * (1.f / 64.f);
            const float ex2 = (asq[v].y + 2.f * c[v] + bsq.y) * (1.f / 64.f);
            const float var = ex2 - mu * mu;
            const float rv  = __builtin_amdgcn_rsqf(var + EPSV) * LOG2E;
            rv_l [r * 16 + m] = rv;
            nmr_l[r * 16 + m] = -mu * rv;
        }
        __syncthreads();   // single-wave WG: s_nop + forced dscnt ordering

        // ---- exp + causal accumulate over the 16x16 pair tile ----
        const bool notdiag = (jt != it);
#pragma unroll
        for (int r = 0; r < 16; ++r) {
            const v2f av = a_val[r];
#pragma unroll
            for (int jj = 0; jj < 16; ++jj) {
                const float rv  = rv_l [r * 16 + jj];  // merged ds_load_b128
                const float nmr = nmr_l[r * 16 + jj];
                const v2f t = av + b_val[jj];          // v_pk_add_f32
                v2f e;
                e.x = __builtin_amdgcn_exp2f(fmaf(t.x, rv, nmr)); // v_fma + v_exp
                e.y = __builtin_amdgcn_exp2f(fmaf(t.y, rv, nmr));
                if (jj <= r || notdiag) {              // compile-time true below diag
                    acc[r] += e;                       // v_pk_add_f32
                }
            }
        }
        __syncthreads();   // protect rv_l / nmr_l before next j-tile
    }

    // ---- Store [B,T,H*D] as float2 per lane ----
    float* Ob = O + h * D;
#pragma unroll
    for (int r = 0; r < 16; ++r) {
        *(v2f*)(Ob + (i0 + r) * RS + 2 * lane) = acc[r];
    }
}

extern "C" void kernel_launch(void* const* d_in, const int* in_sizes, int n_in,
                              void* d_out, int out_size, void* d_ws, size_t ws_size,
                              hipStream_t stream) {
    const float* a = (const float*)d_in[0];
    const float* b = (const float*)d_in[1];
    float* out = (float*)d_out;

    const int H = 8, D = 64;
    const int T = in_sizes[0] / (H * D);   // B=1: T*H*D elements
    const int NT = T / 16;
    const int nrows = T * H;               // memory rows (t*H + h)

    v2f* WA = (v2f*)d_ws;                  // nrows * 8 bytes
    v2f* WB = WA + nrows;                  // nrows * 8 bytes (64 KB total @ T=512)

    row_stats<<<dim3((nrows + 7) / 8), dim3(256), 0, stream>>>(a, b, WA, WB, nrows);
    pairnorm_exp_causal<<<dim3(H * NT), dim3(32), 0, stream>>>(a, b, WA, WB, out, T, NT);
}